// VectorQuantizerPair_42253888258081
// MI455X (gfx1250) — compile-verified
//
#include <hip/hip_runtime.h>
#include <math.h>

#define Gc 8
#define Kc 512
#define Dc 256
#define Bc 8192

typedef __attribute__((ext_vector_type(2))) float v2f;
typedef __attribute__((ext_vector_type(8))) float v8f;

// ---- workspace layout (float units) ----
#define WS_XSQ0 0                      // [B*G]
#define WS_XSQ1 (Bc*Gc)                // [B*G]
#define WS_ESQ  (2*Bc*Gc)              // [G*K]
#define WS_CNT  (2*Bc*Gc + Gc*Kc)      // [2*G*K]  (atomic histograms)
#define WS_SSE  (WS_CNT + 2*Gc*Kc)     // [2*G]    (atomic masked SSE)
#define WS_SHC  (WS_SSE + 16)          // [G]      (atomic shared-row counts)
#define WS_IDX  (WS_SHC + 8)           // [2*G*B]  ints

// ---- d_out layout (flat, return order) ----
constexpr size_t QUANT_SZ  = (size_t)2 * Bc * Gc * Dc;      // 33,554,432
constexpr size_t OFF_QUANT = 2;                              // after 2 scalars
constexpr size_t OFF_PERP  = OFF_QUANT + QUANT_SZ;
constexpr size_t OFF_ENC   = OFF_PERP + 1;
constexpr size_t OFF_OH    = OFF_ENC + QUANT_SZ;             // [G,2,B,K]
constexpr size_t DSZ       = (size_t)Gc * Bc * Kc;           // 33,554,432
constexpr size_t OFF_D0    = OFF_OH + (size_t)Gc * 2 * Bc * Kc;

// Zero the atomic accumulator region of the workspace each call.
__global__ void zero_kernel(float* __restrict__ ws) {
  int t = blockIdx.x * 256 + threadIdx.x;
  int n = WS_IDX - WS_CNT;  // counts + sse + shared-count
  if (t < n) ws[WS_CNT + t] = 0.0f;
}

// Sum-of-squares per row for x0, x1, emb.
__global__ void rowsq_kernel(const float* __restrict__ x0,
                             const float* __restrict__ x1,
                             const float* __restrict__ emb,
                             float* __restrict__ ws) {
  int t = blockIdx.x * blockDim.x + threadIdx.x;
  const float* src;
  float* dst;
  if (t < Bc * Gc) {
    src = x0 + (size_t)t * Dc;               dst = ws + WS_XSQ0 + t;
  } else if (t < 2 * Bc * Gc) {
    int r = t - Bc * Gc;
    src = x1 + (size_t)r * Dc;               dst = ws + WS_XSQ1 + r;
  } else if (t < 2 * Bc * Gc + Gc * Kc) {
    int r = t - 2 * Bc * Gc;
    src = emb + (size_t)r * Dc;              dst = ws + WS_ESQ + r;
  } else {
    return;
  }
  float s = 0.0f;
  for (int d = 0; d < Dc; d += 4) {
    float4 v = *(const float4*)(src + d);
    s += v.x * v.x + v.y * v.y + v.z * v.z + v.w * v.w;
  }
  *dst = s;
}

// WMMA distance kernel, register-blocked: one wave -> 64(b) x 64(k) tile.
// Block = 8 waves covering one 64-row b-strip across all K (8 k-macrotiles),
// so the A slab is shared by all waves of the WGP via WGP$.
// A (16x4 f32): lanes 0-15 rows M, VGPR0/1 = K0/K1; lanes 16-31 = K2/K3.
// B (4x16 f32): symmetric (lane = N column, half selects inner-K pair).
__global__ void __launch_bounds__(256) dist_kernel(const float* __restrict__ x0,
                                                   const float* __restrict__ x1,
                                                   const float* __restrict__ emb,
                                                   const float* __restrict__ ws,
                                                   float* __restrict__ dout) {
  const int z = blockIdx.z;
  const int p = z >> 3;
  const int g = z & 7;
  const float* x   = p ? x1 : x0;
  const float* xsq = ws + (p ? WS_XSQ1 : WS_XSQ0);
  const float* esq = ws + WS_ESQ;

  const int wave = threadIdx.x >> 5;
  const int lane = threadIdx.x & 31;
  const int b0 = blockIdx.x * 64;      // 64-row b macro-tile per block
  const int k0 = wave * 64;            // 64-col k macro-tile per wave (K/64 == 8 waves)
  const int m    = lane & 15;          // row of A / column of B for this lane
  const int half = lane >> 4;          // selects inner-K pair {0,1} vs {2,3}

  const float* ap[4];
  const float* bp[4];
#pragma unroll
  for (int i = 0; i < 4; ++i) {
    ap[i] = x + (size_t)(b0 + i * 16 + m) * (Gc * Dc) + (size_t)g * Dc + half * 2;
    bp[i] = emb + (size_t)g * (Kc * Dc) + (size_t)(k0 + i * 16 + m) * Dc + half * 2;
  }

  v8f acc[4][4] = {};   // [b-subtile][k-subtile]
#pragma unroll 4
  for (int c = 0; c < Dc / 4; ++c) {
    v2f av[4], bv[4];
#pragma unroll
    for (int i = 0; i < 4; ++i) {
      av[i] = *(const v2f*)(ap[i] + 4 * c);
      bv[i] = *(const v2f*)(bp[i] + 4 * c);
    }
#pragma unroll
    for (int bi = 0; bi < 4; ++bi)
#pragma unroll
      for (int ki = 0; ki < 4; ++ki)
        acc[bi][ki] = __builtin_amdgcn_wmma_f32_16x16x4_f32(
            false, av[bi], false, bv[ki], (short)0, acc[bi][ki], false, false);
  }

  // Hoist per-lane row sums (reused across the 4 k-subtiles).
  float xs[4][8];
#pragma unroll
  for (int bi = 0; bi < 4; ++bi)
#pragma unroll
    for (int r = 0; r < 8; ++r)
      xs[bi][r] = xsq[(size_t)(b0 + bi * 16 + half * 8 + r) * Gc + g];

  float* outd = dout + OFF_D0 + (size_t)p * DSZ + (size_t)g * ((size_t)Bc * Kc);
#pragma unroll
  for (int ki = 0; ki < 4; ++ki) {
    const int k = k0 + ki * 16 + m;
    const float es = esq[g * Kc + k];
#pragma unroll
    for (int bi = 0; bi < 4; ++bi) {
#pragma unroll
      for (int r = 0; r < 8; ++r) {
        int b = b0 + bi * 16 + half * 8 + r;   // C layout: M = r (+8 upper half)
        outd[(size_t)b * Kc + k] = xs[bi][r] + es - 2.0f * acc[bi][ki][r];
      }
    }
  }
}

// One wave per (p,g,b): argmin over K, shared-label override, one-hot write,
// count histogram + shared-count atomics, index stash.
__global__ void __launch_bounds__(256) argmin_kernel(const int* __restrict__ labels,
                                                     float* __restrict__ dout,
                                                     float* __restrict__ ws) {
  const int wave = threadIdx.x >> 5, lane = threadIdx.x & 31;
  const int b = blockIdx.x * 8 + wave;
  const int g = blockIdx.y;
  const int p = blockIdx.z;

  const float* drow = dout + OFF_D0 + (size_t)p * DSZ +
                      (size_t)g * ((size_t)Bc * Kc) + (size_t)b * Kc;
  float best = 3.402823466e38f;
  int bestk = 0;
  for (int k = lane; k < Kc; k += 32) {
    float v = drow[k];
    if (v < best) { best = v; bestk = k; }
  }
  for (int off = 16; off > 0; off >>= 1) {
    float ob = __shfl_xor(best, off, 32);
    int   ok = __shfl_xor(bestk, off, 32);
    if (ob < best || (ob == best && ok < bestk)) { best = ob; bestk = ok; }
  }

  const int lab = labels[g * Bc + b];
  const int idx = (lab > -1) ? lab : bestk;

  float* oh = dout + OFF_OH + (size_t)g * (2ull * Bc * Kc) +
              (size_t)p * ((size_t)Bc * Kc) + (size_t)b * Kc;
  for (int k = lane; k < Kc; k += 32) oh[k] = (k == idx) ? 1.0f : 0.0f;

  if (lane == 0) {
    ((int*)(ws + WS_IDX))[(size_t)p * Gc * Bc + (size_t)g * Bc + b] = idx;
    atomicAdd(ws + WS_CNT + p * Gc * Kc + g * Kc + idx, 1.0f);
    if (p == 0 && lab > -1) atomicAdd(ws + WS_SHC + g, 1.0f);
  }
}

// One wave per (p,g,b): gather codeword, write enc_embeddings + straight-through
// quantized, accumulate masked SSE per group.
__global__ void __launch_bounds__(256) gather_kernel(const float* __restrict__ x0,
                                                     const float* __restrict__ x1,
                                                     const float* __restrict__ emb,
                                                     const int* __restrict__ labels,
                                                     float* __restrict__ dout,
                                                     float* __restrict__ ws) {
  const int wave = threadIdx.x >> 5, lane = threadIdx.x & 31;
  const int b = blockIdx.x * 8 + wave;
  const int g = blockIdx.y;
  const int p = blockIdx.z;
  const float* x = p ? x1 : x0;

  const int idx = ((const int*)(ws + WS_IDX))[(size_t)p * Gc * Bc + (size_t)g * Bc + b];
  const int lab = labels[g * Bc + b];

  const float* q  = emb + (size_t)g * Kc * Dc + (size_t)idx * Dc;
  const float* xr = x + (size_t)b * Gc * Dc + (size_t)g * Dc;
  float* enc = dout + OFF_ENC + (size_t)p * Gc * Bc * Dc +
               (size_t)g * Bc * Dc + (size_t)b * Dc;
  float* qt  = dout + OFF_QUANT + (size_t)p * Bc * Gc * Dc +
               (size_t)b * Gc * Dc + (size_t)g * Dc;

  float local = 0.0f;
  for (int d = lane * 4; d < Dc; d += 128) {
    float4 qv = *(const float4*)(q + d);
    float4 xv = *(const float4*)(xr + d);
    *(float4*)(enc + d) = qv;
    float4 st;
    st.x = xv.x + (qv.x - xv.x);
    st.y = xv.y + (qv.y - xv.y);
    st.z = xv.z + (qv.z - xv.z);
    st.w = xv.w + (qv.w - xv.w);
    *(float4*)(qt + d) = st;
    float dx = qv.x - xv.x, dy = qv.y - xv.y, dz = qv.z - xv.z, dw = qv.w - xv.w;
    local += dx * dx + dy * dy + dz * dz + dw * dw;
  }
  for (int off = 16; off > 0; off >>= 1) local += __shfl_xor(local, off, 32);
  if (lane == 0 && lab > -1) atomicAdd(ws + WS_SSE + p * Gc + g, local);
}

// Single wave: scalars (vq loss, pair loss = 0, perplexity).
__global__ void finalize_kernel(float* __restrict__ dout, const float* __restrict__ ws) {
  const int lane = threadIdx.x & 31;
  float perp = 0.0f;
  if (lane < 16) {
    int p = lane >> 3, g = lane & 7;
    const float* cnt = ws + WS_CNT + p * Gc * Kc + g * Kc;
    float s = 0.0f;
    for (int k = 0; k < Kc; ++k) {
      float pr = cnt[k] * (1.0f / (float)Bc);
      s += pr * logf(pr + 1e-10f);
    }
    perp = expf(-s);
  }
  float vq = 0.0f;
  if (lane < Gc) {
    float denom = ws[WS_SHC + lane] * (float)Dc;
    vq = 0.625f * (ws[WS_SSE + lane] + ws[WS_SSE + Gc + lane]) / denom;
  }
  for (int off = 16; off > 0; off >>= 1) {
    perp += __shfl_xor(perp, off, 32);
    vq   += __shfl_xor(vq, off, 32);
  }
  if (lane == 0) {
    dout[0] = vq / (float)Gc;       // avg_vq_loss
    dout[1] = 0.0f;                 // avg_pair_loss
    dout[OFF_PERP] = perp * 0.5f / (float)Gc;  // avg_perplexity
  }
}

extern "C" void kernel_launch(void* const* d_in, const int* in_sizes, int n_in,
                              void* d_out, int out_size, void* d_ws, size_t ws_size,
                              hipStream_t stream) {
  const float* x0     = (const float*)d_in[0];
  const float* x1     = (const float*)d_in[1];
  const float* emb    = (const float*)d_in[2];
  const int*   labels = (const int*)d_in[3];
  float* out = (float*)d_out;
  float* ws  = (float*)d_ws;

  zero_kernel<<<33, 256, 0, stream>>>(ws);

  const int nrows = 2 * Bc * Gc + Gc * Kc;  // 135168, multiple of 256
  rowsq_kernel<<<nrows / 256, 256, 0, stream>>>(x0, x1, emb, ws);

  // 8 waves/block; block = 64-row b-strip, wave w = 64-col k-macrotile.
  dist_kernel<<<dim3(Bc / 64, 1, 16), 256, 0, stream>>>(x0, x1, emb, ws, out);

  argmin_kernel<<<dim3(Bc / 8, Gc, 2), 256, 0, stream>>>(labels, out, ws);
  gather_kernel<<<dim3(Bc / 8, Gc, 2), 256, 0, stream>>>(x0, x1, emb, labels, out, ws);
  finalize_kernel<<<1, 32, 0, stream>>>(out, ws);
}